// PostSegmentationProcessingLayer_85435489452107
// MI455X (gfx1250) — compile-verified
//
#include <hip/hip_runtime.h>
#include <stdint.h>

#define G 384
#define WPL 12                    // 32-bit words per z-line (384/32)
#define NLINES (G * G)            // 147456 (x,y) lines
#define GRID_WORDS (NLINES * WPL) // 1,769,472 u32 per bit-grid (~6.75 MB)

typedef __attribute__((ext_vector_type(2))) float v2f;
typedef __attribute__((ext_vector_type(8))) float v8f;

// Normalized gaussian taps, sigma=0.5, radius 2: exp(-2*t^2)/sum
__device__ __constant__ float GW[5] = {
    2.63865e-4f, 1.06450774e-1f, 7.86570728e-1f, 1.06450774e-1f, 2.63865e-4f};

__device__ __forceinline__ int reflect_idx(int i, int n) {
  if (i < 0) i = -i - 1;
  if (i >= n) i = 2 * n - 1 - i;
  return i;
}

// ---------------------------------------------------------------- clear
__global__ void k_clear(uint32_t* __restrict__ p, int n4) {
  int i = blockIdx.x * blockDim.x + threadIdx.x;
  if (i < n4) ((uint4*)p)[i] = make_uint4(0u, 0u, 0u, 0u);
}

// --------------------------------------------- smooth (WMMA) + scatter
// One wave handles a tile of 16 output rows. Row-axis gaussian as
// Y(16x3) = W(16x20) * Xwin(20x3), done as 5 chained v_wmma_f32_16x16x4_f32.
// Column-axis gaussian (3 cols, symmetric pad) folded in as a 3x3 mix after.
__global__ void __launch_bounds__(256)
k_smooth_scatter(const float* __restrict__ pts, uint32_t* __restrict__ grid,
                 int N, int tilesTotal) {
  __shared__ float shz[8][16][3];
  const int lane = threadIdx.x & 31;
  const int wid  = threadIdx.x >> 5;
  const int wgid = blockIdx.x * (blockDim.x >> 5) + wid;
  const int nw   = gridDim.x * (blockDim.x >> 5);
  const int m16  = lane & 15;
  const bool hi  = lane >= 16;

  // A matrix: banded conv weights W[m][j] = GW[j - m + 2]; lane-constant.
  // 32-bit A 16x4 layout: V0 = K0 (lanes 0-15) / K2 (lanes 16-31), V1 = K1/K3.
  v2f A[5];
#pragma unroll
  for (int kb = 0; kb < 5; ++kb) {
    int j0 = kb * 4 + (hi ? 2 : 0);
    int d0 = j0 - m16 + 2;
    int d1 = d0 + 1;
    v2f a;
    a.x = (d0 >= 0 && d0 < 5) ? GW[d0] : 0.0f;
    a.y = (d1 >= 0 && d1 < 5) ? GW[d1] : 0.0f;
    A[kb] = a;
  }

  for (int t = wgid; t < tilesTotal; t += nw) {
    const int r0 = t * 16;
    v8f acc = {};
#pragma unroll
    for (int kb = 0; kb < 5; ++kb) {
      // B 4x16: V0 = K0 (lanes 0-15) / K2 (lanes 16-31), V1 = K1/K3.
      float b0 = 0.0f, b1 = 0.0f;
      if (m16 < 3) {
        int j0  = kb * 4 + (hi ? 2 : 0);
        int ra  = reflect_idx(r0 - 2 + j0, N);
        int rb  = reflect_idx(r0 - 2 + j0 + 1, N);
        b0 = pts[ra * 3 + m16];
        b1 = pts[rb * 3 + m16];
      }
      v2f B; B.x = b0; B.y = b1;
      acc = __builtin_amdgcn_wmma_f32_16x16x4_f32(
          false, A[kb], false, B, (short)0, acc, false, false);
    }

    // D layout: lane<16 holds (M=r, N=lane) in acc[r]; lane>=16 holds (M=8+r).
    if (!hi && m16 < 3) {
#pragma unroll
      for (int r = 0; r < 8; ++r) shz[wid][r][m16] = acc[r];
    }
    if (hi && m16 < 3) {
#pragma unroll
      for (int r = 0; r < 8; ++r) shz[wid][8 + r][m16] = acc[r];
    }
    asm volatile("s_wait_dscnt 0" ::: "memory");  // wave-internal LDS exchange

    if (lane < 16) {
      int row = r0 + lane;
      if (row < N) {
        float y0 = shz[wid][lane][0];
        float y1 = shz[wid][lane][1];
        float y2 = shz[wid][lane][2];
        // symmetric-pad gaussian over the 3 columns -> 3x3 mix
        float z0 = 0.893021502f * y0 + 0.106714639f * y1 + 0.000263865f * y2;
        float z1 = 0.106714639f * y0 + 0.786570728f * y1 + 0.106714639f * y2;
        float z2 = 0.000263865f * y0 + 0.106714639f * y1 + 0.893021502f * y2;
        int ix = min(G - 1, max(0, (int)floorf(z0 * (float)G)));
        int iy = min(G - 1, max(0, (int)floorf(z1 * (float)G)));
        int iz = min(G - 1, max(0, (int)floorf(z2 * (float)G)));
        atomicOr(&grid[(ix * G + iy) * WPL + (iz >> 5)], 1u << (iz & 31));
      }
    }
    asm volatile("" ::: "memory");
  }
}

// ------------------------------------------------------------- dilation
// Bitwise 3x3x3 max-pool (zero pad): OR of 9 neighbor lines + z shifts.
// Stages 18x18 lines into LDS via async global->LDS copies (ASYNCcnt path).
__global__ void __launch_bounds__(256)
k_dilate(const uint32_t* __restrict__ src, uint32_t* __restrict__ dst) {
  __shared__ uint32_t sh[18 * 18 * WPL];  // 15552 B
  const int tid = threadIdx.x;
  const int gx0 = blockIdx.x * 16, gy0 = blockIdx.y * 16;

  for (int i = tid; i < 18 * 18 * WPL; i += 256) sh[i] = 0u;
  __syncthreads();

  const uint32_t lds_base = (uint32_t)(uintptr_t)(void*)sh;
  for (int t = tid; t < 18 * 18 * 3; t += 256) {
    int li = t / 3, seg = t % 3;
    int lx = li / 18, ly = li % 18;
    int gx = gx0 + lx - 1, gy = gy0 + ly - 1;
    if (gx >= 0 && gx < G && gy >= 0 && gy < G) {
      uint32_t loff = lds_base + (uint32_t)((li * WPL + seg * 4) * 4);
      unsigned long long ga =
          (unsigned long long)(uintptr_t)(src + ((gx * G + gy) * WPL + seg * 4));
      asm volatile("global_load_async_to_lds_b128 %0, %1, off"
                   :: "v"(loff), "v"(ga) : "memory");
    }
  }
  asm volatile("s_wait_asynccnt 0" ::: "memory");
  __syncthreads();

  const int tx = tid >> 4, ty = tid & 15;
  uint32_t L[WPL];
#pragma unroll
  for (int w = 0; w < WPL; ++w) {
    uint32_t v = 0u;
#pragma unroll
    for (int dx = 0; dx < 3; ++dx)
#pragma unroll
      for (int dy = 0; dy < 3; ++dy)
        v |= sh[((tx + dx) * 18 + (ty + dy)) * WPL + w];
    L[w] = v;
  }
  uint32_t O[WPL];
#pragma unroll
  for (int w = 0; w < WPL; ++w) {
    uint32_t lo = (w > 0) ? L[w - 1] : 0u;
    uint32_t hw = (w < WPL - 1) ? L[w + 1] : 0u;
    O[w] = L[w] | (L[w] << 1) | (lo >> 31) | (L[w] >> 1) | (hw << 31);
  }
  uint4* o4 = (uint4*)(dst + ((gx0 + tx) * G + (gy0 + ty)) * WPL);
  o4[0] = make_uint4(O[0], O[1], O[2], O[3]);
  o4[1] = make_uint4(O[4], O[5], O[6], O[7]);
  o4[2] = make_uint4(O[8], O[9], O[10], O[11]);
}

// ------------------------------------------ erosion + unpack to float32
// Bitwise 3x3x3 min-pool (zero pad): AND of 9 lines + z shifts (zeros shift
// in -> borders erode to 0, matching zero-pad min). Writes d_out directly.
__global__ void __launch_bounds__(256)
k_erode_unpack(const uint32_t* __restrict__ src, float* __restrict__ out) {
  __shared__ uint32_t sh[18 * 18 * WPL];
  const int tid = threadIdx.x;
  const int gx0 = blockIdx.x * 16, gy0 = blockIdx.y * 16;

  for (int i = tid; i < 18 * 18 * WPL; i += 256) sh[i] = 0u;
  __syncthreads();

  const uint32_t lds_base = (uint32_t)(uintptr_t)(void*)sh;
  for (int t = tid; t < 18 * 18 * 3; t += 256) {
    int li = t / 3, seg = t % 3;
    int lx = li / 18, ly = li % 18;
    int gx = gx0 + lx - 1, gy = gy0 + ly - 1;
    if (gx >= 0 && gx < G && gy >= 0 && gy < G) {
      uint32_t loff = lds_base + (uint32_t)((li * WPL + seg * 4) * 4);
      unsigned long long ga =
          (unsigned long long)(uintptr_t)(src + ((gx * G + gy) * WPL + seg * 4));
      asm volatile("global_load_async_to_lds_b128 %0, %1, off"
                   :: "v"(loff), "v"(ga) : "memory");
    }
  }
  asm volatile("s_wait_asynccnt 0" ::: "memory");
  __syncthreads();

  const int tx = tid >> 4, ty = tid & 15;
  uint32_t L[WPL];
#pragma unroll
  for (int w = 0; w < WPL; ++w) {
    uint32_t v = 0xFFFFFFFFu;
#pragma unroll
    for (int dx = 0; dx < 3; ++dx)
#pragma unroll
      for (int dy = 0; dy < 3; ++dy)
        v &= sh[((tx + dx) * 18 + (ty + dy)) * WPL + w];
    L[w] = v;
  }
  float4* o4 = (float4*)(out + (size_t)((gx0 + tx) * G + (gy0 + ty)) * G);
  for (int w = 0; w < WPL; ++w) {
    uint32_t lo = (w > 0) ? L[w - 1] : 0u;
    uint32_t hw = (w < WPL - 1) ? L[w + 1] : 0u;
    uint32_t e = L[w] & ((L[w] << 1) | (lo >> 31)) & ((L[w] >> 1) | (hw << 31));
#pragma unroll
    for (int q = 0; q < 8; ++q) {
      o4[w * 8 + q] = make_float4(
          (e >> (q * 4 + 0)) & 1u ? 1.0f : 0.0f,
          (e >> (q * 4 + 1)) & 1u ? 1.0f : 0.0f,
          (e >> (q * 4 + 2)) & 1u ? 1.0f : 0.0f,
          (e >> (q * 4 + 3)) & 1u ? 1.0f : 0.0f);
    }
  }
}

// ---------------------------------------------------------------- launch
extern "C" void kernel_launch(void* const* d_in, const int* in_sizes, int n_in,
                              void* d_out, int out_size, void* d_ws,
                              size_t ws_size, hipStream_t stream) {
  (void)n_in; (void)out_size; (void)ws_size;
  const float* pts = (const float*)d_in[0];
  const int N = in_sizes[0] / 3;

  uint32_t* gridb = (uint32_t*)d_ws;          // 6.75 MB bit grid
  uint32_t* dilb  = gridb + GRID_WORDS;       // 6.75 MB dilated bit grid

  const int n4 = (2 * GRID_WORDS) / 4;
  k_clear<<<(n4 + 255) / 256, 256, 0, stream>>>(gridb, n4);

  const int tiles = (N + 15) / 16;
  k_smooth_scatter<<<256, 256, 0, stream>>>(pts, gridb, N, tiles);

  dim3 mg(G / 16, G / 16);
  k_dilate<<<mg, 256, 0, stream>>>(gridb, dilb);
  k_erode_unpack<<<mg, 256, 0, stream>>>(dilb, (float*)d_out);
}